// DirectionalDiagram_26972394619328
// MI455X (gfx1250) — compile-verified
//
#include <hip/hip_runtime.h>

// D = A x B + C via V_WMMA_F32_16X16X4_F32:
//   A(16x4): col0 = X[m], col1 = 1, cols 2-3 = 0
//   B(4x16): row0 = 1, row1 = Y[n], rows 2-3 = 0
//   C      = 16x16 tile of the input image x
// => D[m,n] = X[m] + Y[n] + x[m,n], i.e. the whole reference computation
//    for one (filter, tile) in a single matrix op.

typedef __attribute__((ext_vector_type(2))) float v2f;
typedef __attribute__((ext_vector_type(8))) float v8f;

#define IMG_W 1024
#define FILTER_COUNT 64
#define TILE 16
#define TILES_PER_ROW (IMG_W / TILE)                       // 64
#define WAVES_PER_BLOCK 8
#define FILTERS_PER_WAVE (FILTER_COUNT / WAVES_PER_BLOCK)  // 8

__global__ __launch_bounds__(256) void dir_diagram_wmma(
    const float* __restrict__ x,
    const float* __restrict__ filters,
    float* __restrict__ out)
{
    const int lane = threadIdx.x & 31;
    const int wave = threadIdx.x >> 5;

    // One 16x16 output tile per block; wave handles FILTERS_PER_WAVE filters.
    const int tile = blockIdx.x;
    const int ti = tile >> 6;      // / TILES_PER_ROW
    const int tj = tile & 63;      // % TILES_PER_ROW
    const int i0 = ti * TILE;
    const int j0 = tj * TILE;

    const int n = lane & 15;       // column within tile (also A-row id)
    const int h = lane >> 4;       // lane half (selects K-pair in A/B, M+8 in C/D)

    // centered, normalized coordinate: (t - (W/2 - 0.5)) / (W * sqrt(2))
    const float scale = 1.0f / (1024.0f * 1.41421356237309515f);
    const float idxA = ((float)(i0 + n) - 511.5f) * scale;  // row coord for A lanes 0-15
    const float idxB = ((float)(j0 + n) - 511.5f) * scale;  // col coord for B lanes 0-15

    // Load C = x tile in WMMA C/D layout:
    //   VGPR r: lanes 0-15 -> x[i0+r][j0+n], lanes 16-31 -> x[i0+r+8][j0+n]
    // Regular (cached) loads: x is 4 MB and reused 64x -> keep it in L2.
    v8f cmat;
    const float* xrow = x + (size_t)(i0 + 8 * h) * IMG_W + (j0 + n);
#pragma unroll
    for (int r = 0; r < 8; ++r)
        cmat[r] = xrow[(size_t)r * IMG_W];

    const int f0 = wave * FILTERS_PER_WAVE;
    float* obase = out + (size_t)f0 * IMG_W * IMG_W
                       + (size_t)(i0 + 8 * h) * IMG_W + (j0 + n);

#pragma unroll
    for (int k = 0; k < FILTERS_PER_WAVE; ++k) {
        const float ang = filters[f0 + k];          // uniform -> scalar load
        const float c = __cosf(ang);
        const float s = __sinf(ang);
        const float Xv = (c - idxA) * c * 0.5f;     // X[f, i0+n]
        const float Yv = (s - idxB) * s * 0.5f;     // Y[f, j0+n]

        // A 16x4 layout: lanes 0-15 v0=K0,v1=K1 ; lanes 16-31 v0=K2,v1=K3
        v2f a, b;
        a.x = (h == 0) ? Xv   : 0.0f;   // A[:,0] = X
        a.y = (h == 0) ? 1.0f : 0.0f;   // A[:,1] = 1
        // B 4x16 layout: lanes 0-15 v0=K0,v1=K1 ; lanes 16-31 v0=K2,v1=K3
        b.x = (h == 0) ? 1.0f : 0.0f;   // B[0,:] = 1
        b.y = (h == 0) ? Yv   : 0.0f;   // B[1,:] = Y

        // 8 args: (neg_a, A, neg_b, B, c_mod, C, reuse_a, reuse_b)
        v8f d = __builtin_amdgcn_wmma_f32_16x16x4_f32(
            false, a, false, b, (short)0, cmat, false, false);

        // Stream 256 MB of output: non-temporal so it doesn't evict x from L2.
        float* o = obase + (size_t)k * IMG_W * IMG_W;
#pragma unroll
        for (int r = 0; r < 8; ++r)
            __builtin_nontemporal_store(d[r], o + (size_t)r * IMG_W);
    }
}

extern "C" void kernel_launch(void* const* d_in, const int* in_sizes, int n_in,
                              void* d_out, int out_size, void* d_ws, size_t ws_size,
                              hipStream_t stream) {
    const float* x       = (const float*)d_in[0];
    const float* filters = (const float*)d_in[1];
    float* out           = (float*)d_out;

    dim3 grid(TILES_PER_ROW * TILES_PER_ROW);  // 4096 tiles
    dim3 block(256);                           // 8 waves: one per 8-filter group
    hipLaunchKernelGGL(dir_diagram_wmma, grid, block, 0, stream, x, filters, out);
}